// NoArbitrageRegularizer_4535485465149
// MI455X (gfx1250) — compile-verified
//
#include <hip/hip_runtime.h>
#include <math.h>

#define EPSF 1e-6f
#define BIGF 1e30f
#define MAXD 16

typedef __attribute__((ext_vector_type(2))) float v2f;
typedef __attribute__((ext_vector_type(8))) float v8f;

// ---------------------------------------------------------------------------
// Per-node precompute: lm/tenor/cp split, liquidity weight, total variance.
// ---------------------------------------------------------------------------
__global__ void k_setup(const float* __restrict__ geo, const float* __restrict__ liq,
                        const float* __restrict__ iv,
                        float* __restrict__ lm, float* __restrict__ tn,
                        float* __restrict__ cpv, float* __restrict__ lw,
                        float* __restrict__ tv, int* __restrict__ nbrcnt, int N) {
  int i = blockIdx.x * blockDim.x + threadIdx.x;
  if (i >= N) return;
  float l = geo[3 * i], t = geo[3 * i + 1], c = geo[3 * i + 2];
  lm[i] = l; tn[i] = t; cpv[i] = c;
  lw[i] = 1.0f / (1.0f + expf(-liq[4 * i + 3]));
  float v = iv[i];
  tv[i] = v * v * fmaxf(t, EPSF);
  nbrcnt[i] = 0;
}

// ---------------------------------------------------------------------------
// Undirected neighbor append (owners = [src,dst], nbrs = [dst,src]).
// Slot order is nondeterministic but canonicalized by the key sort below;
// only entries with distinct finite keys ever reach the masked sums.
// ---------------------------------------------------------------------------
__global__ void k_append(const int* __restrict__ eis, int* __restrict__ nbr,
                         int* __restrict__ nbrcnt, int Es) {
  int t = blockIdx.x * blockDim.x + threadIdx.x;
  if (t >= 2 * Es) return;
  int owner = eis[t];                                  // concat(src,dst) == eis[t]
  int nb = (t < Es) ? eis[Es + t] : eis[t - Es];       // concat(dst,src)
  int slot = atomicAdd(&nbrcnt[owner], 1);
  if (slot < MAXD) nbr[owner * MAXD + slot] = nb;
}

// ---------------------------------------------------------------------------
// Per-node: mask invalid / non-call neighbors (key = +inf), stable insertion
// sort ascending by lm key, emit sorted indices + validity flags.
// ---------------------------------------------------------------------------
__global__ void k_sort(const int* __restrict__ nbr, const int* __restrict__ nbrcnt,
                       const float* __restrict__ lm, const float* __restrict__ cpv,
                       int* __restrict__ ns, int* __restrict__ vs, int N) {
  int i = blockIdx.x * blockDim.x + threadIdx.x;
  if (i >= N) return;
  int cnt = nbrcnt[i]; if (cnt > MAXD) cnt = MAXD;
  float key[MAXD]; int idx[MAXD]; int val[MAXD];
  for (int j = 0; j < MAXD; ++j) {
    int nb = (j < cnt) ? nbr[i * MAXD + j] : -1;
    int safe = nb < 0 ? 0 : nb;
    int v = (nb >= 0) && (cpv[safe] > 0.5f);
    key[j] = v ? lm[safe] : __builtin_inff();
    idx[j] = safe; val[j] = v;
  }
  for (int a = 1; a < MAXD; ++a) {
    float k = key[a]; int id = idx[a]; int v = val[a];
    int b = a - 1;
    while (b >= 0 && key[b] > k) {
      key[b + 1] = key[b]; idx[b + 1] = idx[b]; val[b + 1] = val[b]; --b;
    }
    key[b + 1] = k; idx[b + 1] = id; val[b + 1] = v;
  }
  for (int j = 0; j < MAXD; ++j) {
    ns[i * MAXD + j] = idx[j];
    vs[i * MAXD + j] = val[j];
  }
}

// ---------------------------------------------------------------------------
// WMMA Gram diagonal. One wave per 16-row tile.
// A (16x4 f32): lanes 0-15 hold M=lane, VGPR0=K{0|2}, VGPR1=K{1|3};
// B (4x16 f32): lanes hold N=lane&15, same K striping -> loading E rows with
// the A pattern simultaneously yields A and A^T-as-B, so C += A*A^T (Gram).
//   bestIdx == nullptr : diag[i]      = ||E_i||^2        (+ optional sqrt)
//   bestIdx != nullptr : diag[i]      = E_i . E_best[i]
// ---------------------------------------------------------------------------
__global__ __launch_bounds__(32) void k_gram_diag(const float* __restrict__ E,
                                                  const int* __restrict__ bestIdx,
                                                  float* __restrict__ diag,
                                                  float* __restrict__ diag_sqrt,
                                                  int N, int K) {
  int lane = threadIdx.x;
  int rowBase = blockIdx.x * 16;
  int rowA = rowBase + (lane & 15);
  if (rowA >= N) rowA = N - 1;
  int rowB = bestIdx ? bestIdx[rowA] : rowA;
  const float* pa = E + (size_t)rowA * K;
  const float* pb = E + (size_t)rowB * K;
  int kOff = (lane >> 4) * 2;   // lanes 0-15 -> K {0,1}; lanes 16-31 -> K {2,3}
  v8f c = {};
  for (int k0 = 0; k0 < K; k0 += 4) {
    v2f a, b;
    a.x = pa[k0 + kOff]; a.y = pa[k0 + kOff + 1];
    b.x = pb[k0 + kOff]; b.y = pb[k0 + kOff + 1];
    c = __builtin_amdgcn_wmma_f32_16x16x4_f32(false, a, false, b, (short)0, c,
                                              false, false);
  }
  // C/D layout: VGPR v, lanes 0-15 -> (M=v, N=lane); lanes 16-31 -> (M=v+8).
  // Diagonal m lives at: m<8 -> lane m, c[m];  m>=8 -> lane m+16, c[m-8].
  int dm = (lane < 8) ? lane : ((lane >= 24) ? (lane - 16) : -1);
  if (dm >= 0) {
    int j = dm & 7;
    float dval = 0.f;
#pragma unroll
    for (int t = 0; t < 8; ++t)
      if (t == j) dval = c[t];
    int row = rowBase + dm;
    if (row < N) {
      diag[row] = dval;
      if (diag_sqrt) diag_sqrt[row] = sqrtf(fmaxf(dval, 0.f));
    }
  }
}

// ---------------------------------------------------------------------------
// Nearest-put argmin: one wave per row, lexicographic (dist, index) min
// matching jnp.argmin's first-minimum tie-break.
// ---------------------------------------------------------------------------
__global__ void k_argmin(const float* __restrict__ lm, const float* __restrict__ tn,
                         const float* __restrict__ cpv, int* __restrict__ best, int N) {
  int gid = blockIdx.x * blockDim.x + threadIdx.x;
  int wid = gid >> 5, lane = gid & 31;
  if (wid >= N) return;
  float li = lm[wid], ti = tn[wid];
  float bd = 3.4e38f; int bi = 0x7fffffff;
  for (int j = lane; j < N; j += 32) {
    float dl = li - lm[j], dt = ti - tn[j];
    float d = (cpv[j] < 0.5f) ? (dl * dl + dt * dt) : BIGF;
    if (d < bd) { bd = d; bi = j; }
  }
  for (int off = 16; off >= 1; off >>= 1) {
    float od = __shfl_xor(bd, off, 32);
    int   oi = __shfl_xor(bi, off, 32);
    if (od < bd || (od == bd && oi < bi)) { bd = od; bi = oi; }
  }
  if (lane == 0) best[wid] = bi;
}

// ---------------------------------------------------------------------------
// Calendar penalty: single block, fixed-order tree reduction (deterministic).
// part[0]=sum, part[1]=count
// ---------------------------------------------------------------------------
__global__ void k_calendar(const int* __restrict__ eim, const float* __restrict__ tn,
                           const float* __restrict__ tv, const float* __restrict__ lw,
                           float* __restrict__ part, int Em) {
  __shared__ float ss[256], sc[256];
  int t = threadIdx.x;
  float sum = 0.f, cnt = 0.f;
  for (int e = t; e < Em; e += 256) {
    int ms = eim[e], md = eim[Em + e];
    if (tn[ms] < tn[md] - EPSF) {
      float viol = fmaxf(tv[ms] - tv[md], 0.f);
      sum += fminf(lw[ms], lw[md]) * viol;
      cnt += 1.f;
    }
  }
  ss[t] = sum; sc[t] = cnt; __syncthreads();
  for (int s = 128; s >= 1; s >>= 1) {
    if (t < s) { ss[t] += ss[t + s]; sc[t] += sc[t + s]; }
    __syncthreads();
  }
  if (t == 0) { part[0] = ss[0]; part[1] = sc[0]; }
}

// ---------------------------------------------------------------------------
// Butterfly + embedding-convexity: single block, deterministic reduction.
// part[2]=bf_sum, part[3]=cx_sum, part[4]=trip_count
// ---------------------------------------------------------------------------
__global__ void k_btf(const int* __restrict__ ns, const int* __restrict__ vs,
                      const float* __restrict__ lm, const float* __restrict__ tv,
                      const float* __restrict__ lw, const float* __restrict__ en,
                      const float* __restrict__ cpv, float* __restrict__ part, int N) {
  __shared__ float sb[256], sx[256], st[256];
  int t = threadIdx.x;
  float bsum = 0.f, xsum = 0.f, tcnt = 0.f;
  for (int i = t; i < N; i += 256) {
    if (cpv[i] > 0.5f) {
      float lwi = lw[i];
      for (int j = 0; j + 2 < MAXD; ++j) {
        if (!vs[i * MAXD + j + 2]) continue;   // sorted: valid entries lead
        int n1 = ns[i * MAXD + j], n2 = ns[i * MAXD + j + 1], n3 = ns[i * MAXD + j + 2];
        float lm1 = lm[n1], lm2 = lm[n2], lm3 = lm[n3];
        float denom = lm3 - lm1 + EPSF;
        float alpha = (lm3 - lm2) / denom;
        float beta  = (lm2 - lm1) / denom;
        float wp = lwi * lw[n2];
        float bv = fmaxf(alpha * tv[n1] + beta * tv[n3] - tv[n2], 0.f);
        float xv = fmaxf(en[n2] - (alpha * en[n1] + (1.0f - alpha) * en[n3]), 0.f);
        bsum += wp * bv; xsum += wp * xv; tcnt += 1.f;
      }
    }
  }
  sb[t] = bsum; sx[t] = xsum; st[t] = tcnt; __syncthreads();
  for (int s = 128; s >= 1; s >>= 1) {
    if (t < s) { sb[t] += sb[t + s]; sx[t] += sx[t + s]; st[t] += st[t + s]; }
    __syncthreads();
  }
  if (t == 0) { part[2] = sb[0]; part[3] = sx[0]; part[4] = st[0]; }
}

// ---------------------------------------------------------------------------
// PCP: emb_d = ||E_i||^2 + ||E_b||^2 - 2 E_i.E_b (WMMA-produced terms).
// part[5]=pcp_sum, part[6]=n_calls, part[7]=n_puts
// ---------------------------------------------------------------------------
__global__ void k_pcp(const float* __restrict__ cpv, const float* __restrict__ lw,
                      const float* __restrict__ en2, const float* __restrict__ dotv,
                      const int* __restrict__ best, float* __restrict__ part, int N) {
  __shared__ float sp[256], sc[256], sq[256];
  int t = threadIdx.x;
  float ps = 0.f, nc = 0.f, np = 0.f;
  for (int i = t; i < N; i += 256) {
    float c = cpv[i];
    if (c > 0.5f) {
      int b = best[i];
      float embd = en2[i] + en2[b] - 2.0f * dotv[i];
      ps += lw[i] * lw[b] * embd;
      nc += 1.f;
    }
    if (c < 0.5f) np += 1.f;
  }
  sp[t] = ps; sc[t] = nc; sq[t] = np; __syncthreads();
  for (int s = 128; s >= 1; s >>= 1) {
    if (t < s) { sp[t] += sp[t + s]; sc[t] += sc[t + s]; sq[t] += sq[t + s]; }
    __syncthreads();
  }
  if (t == 0) { part[5] = sp[0]; part[6] = sc[0]; part[7] = sq[0]; }
}

// ---------------------------------------------------------------------------
// Finalize: out = [calendar, butterfly, pcp, convexity, total]
// ---------------------------------------------------------------------------
__global__ void k_final(const float* __restrict__ part, float* __restrict__ out) {
  if (threadIdx.x != 0 || blockIdx.x != 0) return;
  float cal = part[1] > 0.f ? part[0] / part[1] : 0.f;
  float bf  = part[4] > 0.f ? part[2] / part[4] : 0.f;
  float cx  = part[4] > 0.f ? part[3] / part[4] : 0.f;
  float pcp = (part[6] > 0.f && part[7] > 0.f) ? part[5] / part[6] : 0.f;
  out[0] = cal; out[1] = bf; out[2] = pcp; out[3] = cx;
  out[4] = 1.0f * cal + 1.0f * bf + 0.5f * pcp + 0.5f * cx;
}

// ---------------------------------------------------------------------------
extern "C" void kernel_launch(void* const* d_in, const int* in_sizes, int n_in,
                              void* d_out, int out_size, void* d_ws, size_t ws_size,
                              hipStream_t stream) {
  const float* E   = (const float*)d_in[0];   // node_embeddings  N x K
  const float* iv  = (const float*)d_in[1];   // predicted_iv     N x 1
  const float* geo = (const float*)d_in[2];   // geometry         N x 3
  const float* liq = (const float*)d_in[3];   // liquidity        N x 4
  const int*   eis = (const int*)d_in[4];     // edge_index_strike 2 x Es
  const int*   eim = (const int*)d_in[5];     // edge_index_maturity 2 x Em

  const int N  = in_sizes[2] / 3;
  const int K  = in_sizes[0] / N;             // 256
  const int Es = in_sizes[4] / 2;
  const int Em = in_sizes[5] / 2;

  char* w = (char*)d_ws;
  size_t off = 0;
  auto alloc = [&](size_t bytes) -> void* {
    void* p = (void*)(w + off);
    off = (off + bytes + 255) & ~(size_t)255;
    return p;
  };
  float* lm     = (float*)alloc((size_t)N * 4);
  float* tn     = (float*)alloc((size_t)N * 4);
  float* cpv    = (float*)alloc((size_t)N * 4);
  float* lw     = (float*)alloc((size_t)N * 4);
  float* tv     = (float*)alloc((size_t)N * 4);
  float* en2    = (float*)alloc((size_t)N * 4);
  float* en     = (float*)alloc((size_t)N * 4);
  float* dotv   = (float*)alloc((size_t)N * 4);
  int*   best   = (int*)  alloc((size_t)N * 4);
  int*   nbrcnt = (int*)  alloc((size_t)N * 4);
  int*   nbr    = (int*)  alloc((size_t)N * MAXD * 4);
  int*   ns     = (int*)  alloc((size_t)N * MAXD * 4);
  int*   vs     = (int*)  alloc((size_t)N * MAXD * 4);
  float* part   = (float*)alloc(64);

  const int TB = 256;
  const int nbN = (N + TB - 1) / TB;

  k_setup<<<nbN, TB, 0, stream>>>(geo, liq, iv, lm, tn, cpv, lw, tv, nbrcnt, N);
  k_append<<<(2 * Es + TB - 1) / TB, TB, 0, stream>>>(eis, nbr, nbrcnt, Es);
  k_sort<<<nbN, TB, 0, stream>>>(nbr, nbrcnt, lm, cpv, ns, vs, N);
  k_gram_diag<<<(N + 15) / 16, 32, 0, stream>>>(E, nullptr, en2, en, N, K);
  k_argmin<<<((size_t)N * 32 + TB - 1) / TB, TB, 0, stream>>>(lm, tn, cpv, best, N);
  k_gram_diag<<<(N + 15) / 16, 32, 0, stream>>>(E, best, dotv, nullptr, N, K);
  k_calendar<<<1, TB, 0, stream>>>(eim, tn, tv, lw, part, Em);
  k_btf<<<1, TB, 0, stream>>>(ns, vs, lm, tv, lw, en, cpv, part, N);
  k_pcp<<<1, TB, 0, stream>>>(cpv, lw, en2, dotv, best, part, N);
  k_final<<<1, 1, 0, stream>>>(part, (float*)d_out);
}